// GAT_Numbering_Corrector_6828998000940
// MI455X (gfx1250) — compile-verified
//
#include <hip/hip_runtime.h>
#include <hip/hip_bf16.h>
#include <math.h>

// ---------------------------------------------------------------------------
// GATv2 x2 + classifier for MI455X (gfx1250, wave32).
// GEMMs: V_WMMA_F32_16X16X4_F32, branch-free inner loop via zero-padded
// operands (K padded to mult-of-4, N padded to mult-of-16).
// Edge phase: CSR-based, L2-resident float4 gathers, no float atomics.
// ---------------------------------------------------------------------------

#define IN_CH    55
#define IN_PAD   64
#define HID      128
#define HEADS    4
#define DTOT     512
#define OUT_CH   49
#define OUT_PAD  64
#define NEG_SLOPE 0.2f

typedef float v2f __attribute__((ext_vector_type(2)));
typedef float v8f __attribute__((ext_vector_type(8)));

// ---------------------------------------------------------------------------
// Padding / copy helpers (one-shot, tiny)
// ---------------------------------------------------------------------------
__global__ void k_pad_x(const float* __restrict__ x, float* __restrict__ xp, int n) {
    int i = blockIdx.x * blockDim.x + threadIdx.x;          // n*64
    if (i >= n * IN_PAD) return;
    int r = i >> 6, c = i & 63;
    xp[i] = (c < IN_CH) ? x[r * IN_CH + c] : 0.f;
}
__global__ void k_pad_w1(const float* __restrict__ w, float* __restrict__ wp) {
    int i = blockIdx.x * blockDim.x + threadIdx.x;          // 64*512
    if (i >= IN_PAD * DTOT) return;
    int k = i >> 9, c = i & 511;
    wp[i] = (k < IN_CH) ? w[k * DTOT + c] : 0.f;
}
__global__ void k_pad_wcls(const float* __restrict__ w, float* __restrict__ wp) {
    int i = blockIdx.x * blockDim.x + threadIdx.x;          // 512*64
    if (i >= DTOT * OUT_PAD) return;
    int k = i >> 6, c = i & 63;
    wp[i] = (c < OUT_CH) ? w[k * OUT_CH + c] : 0.f;
}
__global__ void k_pad_bcls(const float* __restrict__ b, float* __restrict__ bp) {
    int i = threadIdx.x;                                    // 64
    if (i < OUT_PAD) bp[i] = (i < OUT_CH) ? b[i] : 0.f;
}
__global__ void k_copy_out(const float* __restrict__ op, float* __restrict__ out, int n) {
    int i = blockIdx.x * blockDim.x + threadIdx.x;          // n*49
    if (i >= n * OUT_CH) return;
    int r = i / OUT_CH, c = i - r * OUT_CH;
    out[i] = op[r * OUT_PAD + c];
}

// ---------------------------------------------------------------------------
// CSR construction
// ---------------------------------------------------------------------------
__global__ void k_degree(const int* __restrict__ ei, int E, int N,
                         int* __restrict__ deg) {
    int e = blockIdx.x * blockDim.x + threadIdx.x;
    if (e >= E + N) return;
    int dst = (e < E) ? ei[E + e] : (e - E);
    atomicAdd(&deg[dst], 1);
}

__global__ void k_scan(const int* __restrict__ deg, int* __restrict__ rowptr, int n) {
    __shared__ int sh[1024];
    int t = threadIdx.x;
    int carry = 0;
    for (int base = 0; base < n; base += 1024) {
        int i = base + t;
        int v = (i < n) ? deg[i] : 0;
        sh[t] = v;
        __syncthreads();
        for (int off = 1; off < 1024; off <<= 1) {
            int tv = (t >= off) ? sh[t - off] : 0;
            __syncthreads();
            sh[t] += tv;
            __syncthreads();
        }
        int incl = sh[t];
        if (i < n) rowptr[i] = carry + incl - v;
        int chunk_total = sh[1023];
        __syncthreads();
        carry += chunk_total;
    }
    if (t == 0) rowptr[n] = carry;
}

__global__ void k_fill(const int* __restrict__ ei, int E, int N,
                       const int* __restrict__ rowptr, int* __restrict__ fill,
                       int* __restrict__ csr_src, int* __restrict__ csr_dst) {
    int e = blockIdx.x * blockDim.x + threadIdx.x;
    if (e >= E + N) return;
    int s, d;
    if (e < E) { s = ei[e]; d = ei[E + e]; }
    else       { s = d = e - E; }
    int pos  = atomicAdd(&fill[d], 1);
    int slot = rowptr[d] + pos;
    csr_src[slot] = s;
    csr_dst[slot] = d;
}

// ---------------------------------------------------------------------------
// GEMM: out[M,N] = A[M,K] @ W[K,N] + bias[N]   (fp32 WMMA 16x16x4)
// Preconditions (guaranteed by padding): M%16==0, K%4==0, N%16==0.
// One wave per 16x16 tile; inner loop is branch-free:
//   1x global_load_b64 (A frag) + 2x global_load_b32 (B frag) + 1x v_wmma.
// Tile index is wave-uniform so EXEC is all-ones at every WMMA.
// ---------------------------------------------------------------------------
__global__ void k_gemm_wmma(const float* __restrict__ A,
                            const float* __restrict__ W,
                            const float* __restrict__ bias,
                            float* __restrict__ out,
                            int M, int K, int N) {
    int lane   = threadIdx.x & 31;
    int tile   = blockIdx.x * (blockDim.x >> 5) + (threadIdx.x >> 5);
    int tilesN = N >> 4;
    if (tile >= (M >> 4) * tilesN) return;
    int tm = tile / tilesN;
    int tn = tile - tm * tilesN;

    // A 16x4 frag: lanes 0-15 -> M=lane, K={0,1}; lanes 16-31 -> K={2,3}
    int row  = lane & 15;
    int koff = (lane >> 4) << 1;
    int gn   = tn * 16 + (lane & 15);        // B/D column for this lane

    const float* Arow = A + (size_t)(tm * 16 + row) * K + koff;
    const float* Wcol = W + (size_t)koff * N + gn;
    const size_t wstep = (size_t)N;

    v8f acc = {0.f, 0.f, 0.f, 0.f, 0.f, 0.f, 0.f, 0.f};
    #pragma unroll 4
    for (int kk = 0; kk < K; kk += 4) {
        v2f a = *(const v2f*)(Arow + kk);
        v2f b;
        b.x = Wcol[(size_t)kk * wstep];
        b.y = Wcol[(size_t)kk * wstep + wstep];
        acc = __builtin_amdgcn_wmma_f32_16x16x4_f32(
                  false, a, false, b, (short)0, acc, false, false);
    }

    // D: VGPR r -> lanes 0-15: M=r, N=lane; lanes 16-31: M=8+r, N=lane-16
    float bv    = bias[gn];
    int   mbase = tm * 16 + ((lane < 16) ? 0 : 8);
    #pragma unroll
    for (int r = 0; r < 8; ++r)
        out[(size_t)(mbase + r) * N + gn] = acc[r] + bv;
}

// ---------------------------------------------------------------------------
// Edge scores: one wave per CSR slot; lane owns 16 contiguous channels,
// 8 lanes per head, shfl_xor reduction.
// ---------------------------------------------------------------------------
__global__ void k_edge_scores(const float* __restrict__ xl,
                              const float* __restrict__ xr,
                              const int* __restrict__ csr_src,
                              const int* __restrict__ csr_dst,
                              const float* __restrict__ att,
                              float* __restrict__ scores, int n_slots) {
    int slot = (blockIdx.x * blockDim.x + threadIdx.x) >> 5;
    int lane = threadIdx.x & 31;
    if (slot >= n_slots) return;
    int src = csr_src[slot];
    int dst = csr_dst[slot];
    int h   = lane >> 3;
    int c0  = lane * 16;
    const float4* pl = (const float4*)(xl + (size_t)src * DTOT + c0);
    const float4* pr = (const float4*)(xr + (size_t)dst * DTOT + c0);
    const float4* pa = (const float4*)att + (h * 32 + (lane & 7) * 4);

    float part = 0.f;
    #pragma unroll
    for (int q = 0; q < 4; ++q) {
        float4 a = pl[q], b = pr[q], w = pa[q];
        float ex = a.x + b.x; ex = ex > 0.f ? ex : NEG_SLOPE * ex;
        float ey = a.y + b.y; ey = ey > 0.f ? ey : NEG_SLOPE * ey;
        float ez = a.z + b.z; ez = ez > 0.f ? ez : NEG_SLOPE * ez;
        float ew = a.w + b.w; ew = ew > 0.f ? ew : NEG_SLOPE * ew;
        part += w.x * ex + w.y * ey + w.z * ez + w.w * ew;
    }
    part += __shfl_xor(part, 1);
    part += __shfl_xor(part, 2);
    part += __shfl_xor(part, 4);
    if ((lane & 7) == 0) scores[(size_t)slot * HEADS + h] = part;
}

// ---------------------------------------------------------------------------
// Per-node softmax + aggregation (+bias +ELU). 128 threads/node.
// ---------------------------------------------------------------------------
__global__ void k_aggregate(const float* __restrict__ xl,
                            const float* __restrict__ scores,
                            const int* __restrict__ csr_src,
                            const int* __restrict__ rowptr,
                            const float* __restrict__ bias,
                            float* __restrict__ out, int apply_elu) {
    int node  = blockIdx.x;
    int t     = threadIdx.x;
    int h     = t >> 5;
    int lane  = t & 31;
    int begin = rowptr[node];
    int end   = rowptr[node + 1];

    __shared__ float sm[HEADS];
    __shared__ float sinv[HEADS];

    float m = -3.0e38f;
    for (int j = begin + lane; j < end; j += 32)
        m = fmaxf(m, scores[(size_t)j * HEADS + h]);
    #pragma unroll
    for (int o = 16; o >= 1; o >>= 1) m = fmaxf(m, __shfl_xor(m, o));
    float s = 0.f;
    for (int j = begin + lane; j < end; j += 32)
        s += __expf(scores[(size_t)j * HEADS + h] - m);
    #pragma unroll
    for (int o = 16; o >= 1; o >>= 1) s += __shfl_xor(s, o);
    if (lane == 0) { sm[h] = m; sinv[h] = 1.0f / s; }
    __syncthreads();

    float mh  = sm[h];
    float inv = sinv[h];
    int   c0  = t * 4;
    float4 acc = {0.f, 0.f, 0.f, 0.f};
    for (int j = begin; j < end; ++j) {
        float a = __expf(scores[(size_t)j * HEADS + h] - mh) * inv;
        int  src = csr_src[j];
        float4 v = *(const float4*)(xl + (size_t)src * DTOT + c0);
        acc.x += a * v.x; acc.y += a * v.y;
        acc.z += a * v.z; acc.w += a * v.w;
    }
    float4 bv = *(const float4*)(bias + c0);
    float4 o;
    o.x = acc.x + bv.x; o.y = acc.y + bv.y;
    o.z = acc.z + bv.z; o.w = acc.w + bv.w;
    if (apply_elu) {
        o.x = o.x > 0.f ? o.x : (expf(o.x) - 1.f);
        o.y = o.y > 0.f ? o.y : (expf(o.y) - 1.f);
        o.z = o.z > 0.f ? o.z : (expf(o.z) - 1.f);
        o.w = o.w > 0.f ? o.w : (expf(o.w) - 1.f);
    }
    *(float4*)(out + (size_t)node * DTOT + c0) = o;
}

// ---------------------------------------------------------------------------
// Launch
// ---------------------------------------------------------------------------
extern "C" void kernel_launch(void* const* d_in, const int* in_sizes, int n_in,
                              void* d_out, int out_size, void* d_ws, size_t ws_size,
                              hipStream_t stream) {
    (void)n_in; (void)out_size; (void)ws_size;

    const float* x     = (const float*)d_in[0];
    const int*   ei    = (const int*)  d_in[1];
    const float* w1l   = (const float*)d_in[2];
    const float* b1l   = (const float*)d_in[3];
    const float* w1r   = (const float*)d_in[4];
    const float* b1r   = (const float*)d_in[5];
    const float* att1  = (const float*)d_in[6];
    const float* bias1 = (const float*)d_in[7];
    const float* w2l   = (const float*)d_in[8];
    const float* b2l   = (const float*)d_in[9];
    const float* w2r   = (const float*)d_in[10];
    const float* b2r   = (const float*)d_in[11];
    const float* att2  = (const float*)d_in[12];
    const float* bias2 = (const float*)d_in[13];
    const float* wcls  = (const float*)d_in[14];
    const float* bcls  = (const float*)d_in[15];

    const int N  = in_sizes[0] / IN_CH;   // 20000
    const int E  = in_sizes[1] / 2;       // 320000
    const int Et = E + N;                 // + self loops

    // ---- workspace carve (floats first, then ints; 16B aligned) ----
    const size_t nodeF = (size_t)N * DTOT;
    float* A      = (float*)d_ws;                 // xl (41 MB)
    float* B      = A + nodeF;                    // xr (41 MB)
    float* C      = B + nodeF;                    // h  (41 MB)
    float* sc     = C + nodeF;                    // scores [Et,4]
    float* xpad   = sc + (size_t)Et * HEADS;      // [N,64]
    float* outpad = xpad + (size_t)N * IN_PAD;    // [N,64]
    float* w1lp   = outpad + (size_t)N * OUT_PAD; // [64,512]
    float* w1rp   = w1lp + (size_t)IN_PAD * DTOT;
    float* wclsp  = w1rp + (size_t)IN_PAD * DTOT; // [512,64]
    float* bclsp  = wclsp + (size_t)DTOT * OUT_PAD; // [64]
    int* csr_src  = (int*)(bclsp + OUT_PAD);
    int* csr_dst  = csr_src + Et;
    int* rowptr   = csr_dst + Et;                 // N+1
    int* deg      = rowptr + (N + 1);
    int* fill     = deg + N;

    // ---- operand padding (removes all bounds checks from GEMM hot loop) ----
    k_pad_x   <<<(N * IN_PAD + 255) / 256, 256, 0, stream>>>(x, xpad, N);
    k_pad_w1  <<<(IN_PAD * DTOT + 255) / 256, 256, 0, stream>>>(w1l, w1lp);
    k_pad_w1  <<<(IN_PAD * DTOT + 255) / 256, 256, 0, stream>>>(w1r, w1rp);
    k_pad_wcls<<<(DTOT * OUT_PAD + 255) / 256, 256, 0, stream>>>(wcls, wclsp);
    k_pad_bcls<<<1, 64, 0, stream>>>(bcls, bclsp);

    // ---- CSR build (graph identical for both layers) ----
    hipMemsetAsync(deg, 0, sizeof(int) * (size_t)2 * N, stream);
    int eblocks = (Et + 255) / 256;
    k_degree<<<eblocks, 256, 0, stream>>>(ei, E, N, deg);
    k_scan<<<1, 1024, 0, stream>>>(deg, rowptr, N);
    k_fill<<<eblocks, 256, 0, stream>>>(ei, E, N, rowptr, fill, csr_src, csr_dst);

    // GEMM grids: 8 waves (16x16 tiles) per 256-thread block
    const int tilesM = N / 16;                         // 1250
    const int gblkD  = (tilesM * (DTOT / 16) + 7) / 8; // 5000
    const int gblkC  = (tilesM * (OUT_PAD / 16) + 7) / 8; // 625
    const int sblk   = (Et + 7) / 8;

    // ---- layer 1 ----
    k_gemm_wmma<<<gblkD, 256, 0, stream>>>(xpad, w1lp, b1l, A, N, IN_PAD, DTOT);
    k_gemm_wmma<<<gblkD, 256, 0, stream>>>(xpad, w1rp, b1r, B, N, IN_PAD, DTOT);
    k_edge_scores<<<sblk, 256, 0, stream>>>(A, B, csr_src, csr_dst, att1, sc, Et);
    k_aggregate<<<N, 128, 0, stream>>>(A, sc, csr_src, rowptr, bias1, C, 1);

    // ---- layer 2 ----
    k_gemm_wmma<<<gblkD, 256, 0, stream>>>(C, w2l, b2l, A, N, DTOT, DTOT);
    k_gemm_wmma<<<gblkD, 256, 0, stream>>>(C, w2r, b2r, B, N, DTOT, DTOT);
    k_edge_scores<<<sblk, 256, 0, stream>>>(A, B, csr_src, csr_dst, att2, sc, Et);
    k_aggregate<<<N, 128, 0, stream>>>(A, sc, csr_src, rowptr, bias2, C, 1);

    // ---- classifier (padded N=64, then narrow copy to d_out) ----
    k_gemm_wmma<<<gblkC, 256, 0, stream>>>(C, wclsp, bclsp, outpad, N, DTOT, OUT_PAD);
    k_copy_out<<<(N * OUT_CH + 255) / 256, 256, 0, stream>>>(outpad, (float*)d_out, N);
}